// GraphSageLayer_53755810677325
// MI455X (gfx1250) — compile-verified
//
#include <hip/hip_runtime.h>
#include <stdint.h>

typedef __attribute__((ext_vector_type(2))) float v2f;
typedef __attribute__((ext_vector_type(8))) float v8f;

#define N_NODES 100000
#define CH 128
#define N_EDGES 640000

// ---------------------------------------------------------------------------
// Kernel 1: build concatenated/transposed weights WcT[n][k], k in [0,256):
//   WcT[n][k] = (k < 128) ? W_l[n][k] : W_r[n][k-128]
// so the WMMA B-fragment (column n of the combined weight) is a contiguous row.
// ---------------------------------------------------------------------------
__global__ void build_wct(const float* __restrict__ Wl,
                          const float* __restrict__ Wr,
                          float* __restrict__ WcT) {
    int t = blockIdx.x * blockDim.x + threadIdx.x;   // 0 .. 128*256-1
    int n = t >> 8;
    int k = t & 255;
    float v = (k < CH) ? Wl[n * CH + k] : Wr[n * CH + (k - CH)];
    WcT[n * 256 + k] = v;
}

// ---------------------------------------------------------------------------
// Kernel 2: edge scatter-add.  One 32-lane group per edge; each lane handles
// 4 channels (float4 coalesced gather + 4 fp32 global atomics into agg[dst]).
// Lane 0 also counts the edge for the degree.
// ---------------------------------------------------------------------------
__global__ void edge_scatter(const float* __restrict__ x,
                             const long long* __restrict__ ei,   // int64 [2, E]
                             float* __restrict__ agg,
                             unsigned int* __restrict__ cnt) {
    long long gid = (long long)blockIdx.x * blockDim.x + threadIdx.x;
    int e    = (int)(gid >> 5);
    int lane = (int)(gid & 31);
    if (e >= N_EDGES) return;

    int src = (int)ei[e];
    int dst = (int)ei[N_EDGES + e];

    const float4 v = *(const float4*)(x + (size_t)src * CH + lane * 4);
    float* base = agg + (size_t)dst * CH + lane * 4;
    unsafeAtomicAdd(base + 0, v.x);
    unsafeAtomicAdd(base + 1, v.y);
    unsafeAtomicAdd(base + 2, v.z);
    unsafeAtomicAdd(base + 3, v.w);
    if (lane == 0) atomicAdd(&cnt[dst], 1u);
}

// ---------------------------------------------------------------------------
// Kernel 3: fused GEMM  out = (agg/deg) @ W_l^T + b_l + x @ W_r^T
// One wave per 16x16 output tile, K = 256 in steps of 4 via
// v_wmma_f32_16x16x4_f32 (full fp32 matrix core).
//
// fp32 WMMA operand layout (wave32):
//   A (16x4):  lanes 0-15 -> M=lane, regs {K0,K1}; lanes 16-31 -> {K2,K3}
//   B (4x16):  lanes 0-15 -> N=lane, regs {K0,K1}; lanes 16-31 -> {K2,K3}
//   C/D:       reg v, lanes 0-15 -> M=v, N=lane; lanes 16-31 -> M=8+v
// ---------------------------------------------------------------------------
__global__ __launch_bounds__(256) void sage_gemm(
    const float* __restrict__ x,
    const float* __restrict__ agg,
    const unsigned int* __restrict__ cnt,
    const float* __restrict__ WcT,      // [128][256]
    const float* __restrict__ bl,       // [128]
    float* __restrict__ out) {

    const int wave = threadIdx.x >> 5;
    const int lane = threadIdx.x & 31;
    const int lr   = lane & 15;
    const int hi   = lane >> 4;              // lane-half selects K pair {0,1} vs {2,3}

    const int row = blockIdx.x * 16 + lr;    // A row supplied by this lane
    const int col = wave * 16 + lr;          // B column supplied by this lane

    const float deg = (float)cnt[row];
    const float inv = 1.0f / fmaxf(deg, 1.0f);   // mean; isolated nodes -> 0 anyway

    const float* arow = agg + (size_t)row * CH;
    const float* xrow = x   + (size_t)row * CH;
    const float* bcol = WcT + (size_t)col * 256;

    v8f acc = {};

    // K = 0..127 : mean-aggregated neighbor features vs W_l
    #pragma unroll
    for (int kk = 0; kk < 32; ++kk) {
        const int kb = kk * 4 + hi * 2;
        v2f a, b;
        a.x = arow[kb]     * inv;
        a.y = arow[kb + 1] * inv;
        b.x = bcol[kb];
        b.y = bcol[kb + 1];
        acc = __builtin_amdgcn_wmma_f32_16x16x4_f32(
                  false, a, false, b, (short)0, acc, false, false);
    }
    // K = 128..255 : self features vs W_r
    #pragma unroll
    for (int kk = 0; kk < 32; ++kk) {
        const int kb = kk * 4 + hi * 2;
        v2f a, b;
        a.x = xrow[kb];
        a.y = xrow[kb + 1];
        b.x = bcol[CH + kb];
        b.y = bcol[CH + kb + 1];
        acc = __builtin_amdgcn_wmma_f32_16x16x4_f32(
                  false, a, false, b, (short)0, acc, false, false);
    }

    // Store D with bias: this lane owns column n = col, rows m = v + 8*hi.
    const float bias = bl[col];
    float* orow = out + (size_t)(blockIdx.x * 16 + hi * 8) * CH + col;
    #pragma unroll
    for (int v = 0; v < 8; ++v) {
        orow[(size_t)v * CH] = acc[v] + bias;
    }
}

// ---------------------------------------------------------------------------
extern "C" void kernel_launch(void* const* d_in, const int* in_sizes, int n_in,
                              void* d_out, int out_size, void* d_ws, size_t ws_size,
                              hipStream_t stream) {
    const float*     x  = (const float*)d_in[0];
    const long long* ei = (const long long*)d_in[1];   // int64 edge_index [2, E]
    const float*     Wl = (const float*)d_in[2];
    const float*     bl = (const float*)d_in[3];
    const float*     Wr = (const float*)d_in[4];
    float* out = (float*)d_out;

    char* ws = (char*)d_ws;
    const size_t agg_bytes = (size_t)N_NODES * CH * sizeof(float);   // 51.2 MB
    const size_t cnt_bytes = (size_t)N_NODES * sizeof(unsigned int); // 0.4 MB
    float*        agg = (float*)ws;
    unsigned int* cnt = (unsigned int*)(ws + agg_bytes);
    float*        WcT = (float*)(ws + agg_bytes + cnt_bytes);        // 128 KB

    // Zero the accumulators every call (harness does not re-poison/zero ws).
    hipMemsetAsync(agg, 0, agg_bytes + cnt_bytes, stream);

    // Pack weights.
    build_wct<<<(CH * 256) / 256, 256, 0, stream>>>(Wl, Wr, WcT);

    // Scatter-add neighbor features + degree counts.
    const long long tot = (long long)N_EDGES * 32;
    const int sblocks = (int)((tot + 255) / 256);
    edge_scatter<<<sblocks, 256, 0, stream>>>(x, ei, agg, cnt);

    // Fused mean-normalize + dual GEMM + bias via fp32 WMMA.
    sage_gemm<<<N_NODES / 16, 256, 0, stream>>>(x, agg, cnt, WcT, bl, out);
}